// Discriminator_14439680049449
// MI455X (gfx1250) — compile-verified
//
#include <hip/hip_runtime.h>
#include <math.h>

typedef __attribute__((ext_vector_type(2))) float v2f;
typedef __attribute__((ext_vector_type(8))) float v8f;

// ---------------------------------------------------------------------------
// Dense part of graph_conv, one 16-vertex tile per wave32, IN_D compile-time:
//   y0 = x @ w0^T + b0  -> out0  (row stride ld_out, column-band base applied)
//   y1 = x @ w1^T + b1  -> y1out (row stride 16)
// V_WMMA_F32_16X16X4_F32; K stepped by 4; one A load feeds both GEMMs.
// ISA 32-bit A/B layout: lane L holds (row/col L%16), VGPR pair = K{k0,k0+1}
// for lanes 0-15 and K{k0+2,k0+3} for lanes 16-31  =>  kk = k0 + 2*(L/16).
// D layout: VGPR r -> row r + 8*(L/16), col L%16.
// Requires numTiles*16 rows valid (V % 16 == 0; V = 1e6 here).
// ---------------------------------------------------------------------------
template<int IN_D>
__global__ __launch_bounds__(256) void gconv_dense_wmma(
    const float* __restrict__ xin, int ldx,
    const float* __restrict__ w0, const float* __restrict__ b0,
    const float* __restrict__ w1, const float* __restrict__ b1,
    float* __restrict__ out0, int ld_out,
    float* __restrict__ y1out,
    int numTiles)
{
  const int wave = (blockIdx.x * blockDim.x + threadIdx.x) >> 5;
  const int lane = threadIdx.x & 31;
  if (wave >= numTiles) return;           // wave-uniform: EXEC stays all-1s

  const int half = lane >> 4;             // 0: lanes 0-15, 1: lanes 16-31
  const int l16  = lane & 15;
  const int vrow = wave * 16 + l16;       // A row (vertex)
  const int ncol = l16;                   // B/C/D column

  // C init = bias broadcast down the column
  v8f c0, c1;
  const float bb0 = b0[ncol];
  const float bb1 = b1[ncol];
#pragma unroll
  for (int r = 0; r < 8; ++r) { c0[r] = bb0; c1[r] = bb1; }

  const float* __restrict__ xrow  = xin + (long)vrow * ldx;
  const float* __restrict__ w0row = w0 + ncol * IN_D;   // row n of w0 [16, IN_D]
  const float* __restrict__ w1row = w1 + ncol * IN_D;

  if constexpr (IN_D % 4 == 0) {
    // All addresses 8B-aligned: ldx=96 rows, IN_D multiple of 16 weight rows.
#pragma unroll
    for (int k0 = 0; k0 < IN_D; k0 += 4) {
      const int kk = k0 + 2 * half;
      const v2f a   = *(const v2f*)(xrow  + kk);
      const v2f bv0 = *(const v2f*)(w0row + kk);
      const v2f bv1 = *(const v2f*)(w1row + kk);
      c0 = __builtin_amdgcn_wmma_f32_16x16x4_f32(false, a, false, bv0, (short)0, c0, false, false);
      c1 = __builtin_amdgcn_wmma_f32_16x16x4_f32(false, a, false, bv1, (short)0, c1, false, false);
    }
  } else {
    // IN_D == 3: one K=4 WMMA with K=3 zero-padded. The pad element is
    // produced by loading a SAFE in-bounds address then selecting 0
    // (v_cndmask), avoiding exec-divergent predicated loads.
    v2f a, bv0, bv1;
    a.x   = xrow [2 * half];              // K=0 (half 0) / K=2 (half 1): in bounds
    bv0.x = w0row[2 * half];
    bv1.x = w1row[2 * half];
    const int oy = half ? 0 : 1;          // safe address; half==1 value discarded
    const float ta  = xrow [oy];
    const float tb0 = w0row[oy];
    const float tb1 = w1row[oy];
    a.y   = half ? 0.f : ta;              // K=1 real, K=3 zero pad
    bv0.y = half ? 0.f : tb0;
    bv1.y = half ? 0.f : tb1;
    c0 = __builtin_amdgcn_wmma_f32_16x16x4_f32(false, a, false, bv0, (short)0, c0, false, false);
    c1 = __builtin_amdgcn_wmma_f32_16x16x4_f32(false, a, false, bv1, (short)0, c1, false, false);
  }

  const int rbase = wave * 16 + half * 8;
#pragma unroll
  for (int r = 0; r < 8; ++r) {
    out0[(long)(rbase + r) * ld_out + ncol] = c0[r];
    y1out[(long)(rbase + r) * 16    + ncol] = c1[r];
  }
}

// ---------------------------------------------------------------------------
// Undirected edge scatter: out[a] += y1[b]; out[b] += y1[a].
// One thread per (edge, 4-feature group): y1 gathers are b128 loads
// (rows 64B-aligned), edge indices loaded 4x less. y1 (64MB) + dest band
// (64MB) are L2-resident on MI455X (192MB L2), so the f32 atomics run near
// L2 bandwidth.
// ---------------------------------------------------------------------------
__global__ __launch_bounds__(256) void edge_scatter(
    const int* __restrict__ edges,
    const float* __restrict__ y1,
    float* __restrict__ out, int ld_out, long total4)
{
  const long idx = (long)blockIdx.x * blockDim.x + threadIdx.x;
  if (idx >= total4) return;
  const long e  = idx >> 2;
  const int  fb = (int)(idx & 3) * 4;
  const int  a  = edges[2 * e];
  const int  b  = edges[2 * e + 1];
  const float4 yb = *(const float4*)(y1 + (long)b * 16 + fb);
  const float4 ya = *(const float4*)(y1 + (long)a * 16 + fb);
  float* oa = out + (long)a * ld_out + fb;
  float* ob = out + (long)b * ld_out + fb;
  atomicAdd(oa + 0, yb.x); atomicAdd(oa + 1, yb.y);
  atomicAdd(oa + 2, yb.z); atomicAdd(oa + 3, yb.w);
  atomicAdd(ob + 0, ya.x); atomicAdd(ob + 1, ya.y);
  atomicAdd(ob + 2, ya.z); atomicAdd(ob + 3, ya.w);
}

// ---------------------------------------------------------------------------
// Column sums of x[V,96]: blockDim = 96, one column per thread, coalesced
// row reads, one atomicAdd per (block, column).
// ---------------------------------------------------------------------------
__global__ void colsum_kernel(const float* __restrict__ x, int V,
                              float* __restrict__ colsum)
{
  const int c = threadIdx.x;          // 0..95
  float s = 0.f;
  for (long v = blockIdx.x; v < V; v += gridDim.x)
    s += x[v * 96 + c];
  atomicAdd(&colsum[c], s);
}

__global__ void zero96(float* __restrict__ p)
{
  if (threadIdx.x < 96) p[threadIdx.x] = 0.f;
}

// out = sigmoid(mean(x) @ fcw^T + fcb)
__global__ void finalize(const float* __restrict__ colsum,
                         const float* __restrict__ fcw,
                         const float* __restrict__ fcb,
                         float invV, float* __restrict__ out)
{
  __shared__ float red[128];
  const int t = threadIdx.x;
  red[t] = (t < 96) ? colsum[t] * invV * fcw[t] : 0.f;
  __syncthreads();
#pragma unroll
  for (int s = 64; s > 0; s >>= 1) {
    if (t < s) red[t] += red[t + s];
    __syncthreads();
  }
  if (t == 0) {
    const float z = red[0] + fcb[0];
    out[0] = 1.f / (1.f + expf(-z));
  }
}

// ---------------------------------------------------------------------------
// d_in order (setup_inputs dict, gparams flattened):
//  0 verts[V,3] f32   1 edges[E,2] i32
//  2 cw0[16,3]  3 cb0[16]  4 cw1[16,3]  5 cb1[16]
//  6..25 gparams: 5 x (w0[16,H], b0[16], w1[16,H], b1[16])
//  26 fcw[1,96]  27 fcb[1]
// ---------------------------------------------------------------------------
extern "C" void kernel_launch(void* const* d_in, const int* in_sizes, int n_in,
                              void* d_out, int out_size, void* d_ws, size_t ws_size,
                              hipStream_t stream)
{
  const float* verts = (const float*)d_in[0];
  const int*   edges = (const int*)  d_in[1];
  const float* cw0   = (const float*)d_in[2];
  const float* cb0   = (const float*)d_in[3];
  const float* cw1   = (const float*)d_in[4];
  const float* cb1   = (const float*)d_in[5];
  const float* fcw   = (const float*)d_in[26];
  const float* fcb   = (const float*)d_in[27];

  const int  V = in_sizes[0] / 3;
  const long E = (long)in_sizes[1] / 2;

  // Workspace layout: x[V,96] | y1[V,16] | colsum[96]
  float* x      = (float*)d_ws;
  float* y1     = x  + (size_t)V * 96;
  float* colsum = y1 + (size_t)V * 16;

  const int numTiles    = V / 16;         // V = 1e6 -> 62500 exact tiles
  const int wavesPerBlk = 256 / 32;
  const int denseBlocks = (numTiles + wavesPerBlk - 1) / wavesPerBlk;

  const long scatTotal4 = E * 4;          // (edge, 4-feature-group) threads
  const int  scatBlocks = (int)((scatTotal4 + 255) / 256);

  // ---- layer 0: verts(3) -> x[:, 0:16] ----
  gconv_dense_wmma<3><<<denseBlocks, 256, 0, stream>>>(
      verts, 3, cw0, cb0, cw1, cb1, x, 96, y1, numTiles);
  edge_scatter<<<scatBlocks, 256, 0, stream>>>(edges, y1, x, 96, scatTotal4);

  // ---- layers 1..5: x[:, 0:H] -> x[:, H:H+16] (concat is free) ----
  // H = 16, 32, 48, 64, 80; output band base = H.
  {
    const float* w0; const float* b0; const float* w1; const float* b1;
#define GLAYER(I, H)                                                         \
    w0 = (const float*)d_in[6 + 4 * (I) + 0];                                \
    b0 = (const float*)d_in[6 + 4 * (I) + 1];                                \
    w1 = (const float*)d_in[6 + 4 * (I) + 2];                                \
    b1 = (const float*)d_in[6 + 4 * (I) + 3];                                \
    gconv_dense_wmma<H><<<denseBlocks, 256, 0, stream>>>(                    \
        x, 96, w0, b0, w1, b1, x + (H), 96, y1, numTiles);                   \
    edge_scatter<<<scatBlocks, 256, 0, stream>>>(edges, y1, x + (H), 96,     \
                                                 scatTotal4);
    GLAYER(0, 16)
    GLAYER(1, 32)
    GLAYER(2, 48)
    GLAYER(3, 64)
    GLAYER(4, 80)
#undef GLAYER
  }

  // ---- head: mean over V, FC 96->1, sigmoid ----
  zero96<<<1, 128, 0, stream>>>(colsum);
  colsum_kernel<<<2048, 96, 0, stream>>>(x, V, colsum);
  finalize<<<1, 128, 0, stream>>>(colsum, fcw, fcb, 1.0f / (float)V, (float*)d_out);
}